// SISOLayer_46462956208717
// MI455X (gfx1250) — compile-verified
//
#include <hip/hip_runtime.h>

// S4-DPLR layer, H=512 heads, L=2048 steps, N=64 states.
// Reformulated (x0==0) as: K[t] = Re(Cb . Ab^t . Bb); y = causal_conv(K,u) + D*u.
// Three kernels, one workgroup (8 wave32) per head:
//   1) discretize: WMMA complex 64x64 matmuls (Ab, 11 squarings), Gauss-Jordan
//   2) kernel-gen: 127 chunk steps B_next = Ab^16 @ B_cur via WMMA; TDM staging
//   3) causal Toeplitz conv via WMMA; TDM staging of K row + strided u column

#define H_DIM 512
#define L_DIM 2048
#define N_DIM 64
#define MPAD  65   // LDS row pitch for 64x64 tiles: 65 % 64 == 1 -> conflict-free

typedef __attribute__((ext_vector_type(2))) float v2f;
typedef __attribute__((ext_vector_type(8))) float v8f;
typedef __attribute__((ext_vector_type(4))) unsigned int u32x4;
typedef __attribute__((ext_vector_type(8))) int i32x8;
typedef __attribute__((ext_vector_type(4))) int i32x4;

__device__ __forceinline__ v8f wmma_f32x4(v2f a, v2f b, v8f c) {
  // (neg_a, A, neg_b, B, c_mod, C, reuse_a, reuse_b); f32 WMMA: A/B neg illegal.
  return __builtin_amdgcn_wmma_f32_16x16x4_f32(false, a, false, b, (short)0, c,
                                               false, false);
}

// ---------------------------------------------------------------------------
// Tensor Data Mover: 2D tile load Global -> LDS (D# per CDNA5 ISA)
// pad_interval code 5 == pad after every 64 DWORDs; pad_amount 0 == 1 DWORD,
// which reproduces the MPAD=65 bank-conflict-free pitch during the DMA.
// ---------------------------------------------------------------------------
#if __has_builtin(__builtin_amdgcn_tensor_load_to_lds)
#define HAVE_TDM 1
#else
#define HAVE_TDM 0
#endif

__device__ __forceinline__ void tdm_load_2d(unsigned lds_addr, const void* gptr,
                                            unsigned tensor_d0, unsigned tile_d0,
                                            unsigned tensor_d1, unsigned tile_d1,
                                            unsigned long long stride0_elems,
                                            int pad_en, int pad_interval_code) {
#if HAVE_TDM
  const unsigned long long ga = (unsigned long long)gptr;
  u32x4 g0;
  g0.x = 1u;                                  // count=1, user descriptor
  g0.y = lds_addr;                            // LDS byte address
  g0.z = (unsigned)ga;                        // global_addr[31:0]
  g0.w = (unsigned)((ga >> 32) & 0x01ffffffu) | (2u << 30);  // [56:32], type=2
  i32x8 g1;
  g1[0] = (2 << 16)                           // data_size = 4 bytes
        | (pad_en << 20) | (pad_interval_code << 22);  // pad_amount=0 (1 DWORD)
  g1[1] = (int)((tensor_d0 & 0xffffu) << 16);            // tensor_dim0[15:0]
  g1[2] = (int)((tensor_d0 >> 16) & 0xffffu)             // tensor_dim0[31:16]
        | (int)((tensor_d1 & 0xffffu) << 16);            // tensor_dim1[15:0]
  g1[3] = (int)((tensor_d1 >> 16) & 0xffffu)             // tensor_dim1[31:16]
        | (int)((tile_d0 & 0xffffu) << 16);              // tile_dim0
  g1[4] = (int)(tile_d1 & 0xffffu);                      // tile_dim1; tile_dim2=0
  g1[5] = (int)(stride0_elems & 0xffffffffu);            // tensor_dim0_stride lo
  g1[6] = (int)((stride0_elems >> 32) & 0xffffu);        // stride hi; dim1_stride=0
  g1[7] = 0;
  const i32x4 gz = {0, 0, 0, 0};
#if __has_include(<hip/amd_detail/amd_gfx1250_TDM.h>)
  const i32x8 gz8 = {0, 0, 0, 0, 0, 0, 0, 0};            // clang-23: 6-arg form
  __builtin_amdgcn_tensor_load_to_lds(g0, g1, gz, gz, gz8, 0);
#else
  __builtin_amdgcn_tensor_load_to_lds(g0, g1, gz, gz, 0); // ROCm7.2: 5-arg form
#endif
#endif
}

// Complex 64x64 matmul D = A*B, all operands in LDS with row pitch MPAD.
// 16 output tiles of 16x16; 8 waves -> 2 tiles per wave. 4 real WMMA
// accumulator chains per tile (rr, ii, ri, ir), combined at writeback.
// Caller must barrier before (operands ready) and after (D complete).
__device__ void cmm64(float* __restrict__ Dre, float* __restrict__ Dim,
                      const float* Are, const float* Aim,
                      const float* Bre, const float* Bim) {
  const int tid  = threadIdx.x;
  const int lane = tid & 31;
  const int wave = tid >> 5;
  const int lo   = lane & 15;
  const int hi   = lane >> 4;
  for (int t = wave; t < 16; t += 8) {
    const int mi = t >> 2, ni = t & 3;
    v8f rr = {}, ii = {}, ri = {}, ir = {};
    for (int kk = 0; kk < 16; ++kk) {
      const int arow = 16 * mi + lo;
      const int acol = 4 * kk + 2 * hi;
      v2f aRe, aIm, bRe, bIm;
      aRe.x = Are[arow * MPAD + acol];     aRe.y = Are[arow * MPAD + acol + 1];
      aIm.x = Aim[arow * MPAD + acol];     aIm.y = Aim[arow * MPAD + acol + 1];
      const int bcol = 16 * ni + lo;
      const int brow = 4 * kk + 2 * hi;
      bRe.x = Bre[brow * MPAD + bcol];     bRe.y = Bre[(brow + 1) * MPAD + bcol];
      bIm.x = Bim[brow * MPAD + bcol];     bIm.y = Bim[(brow + 1) * MPAD + bcol];
      rr = wmma_f32x4(aRe, bRe, rr);
      ii = wmma_f32x4(aIm, bIm, ii);
      ri = wmma_f32x4(aRe, bIm, ri);
      ir = wmma_f32x4(aIm, bRe, ir);
    }
    for (int j = 0; j < 8; ++j) {
      const int row = 16 * mi + j + 8 * hi;
      const int col = 16 * ni + lo;
      Dre[row * MPAD + col] = rr[j] - ii[j];
      Dim[row * MPAD + col] = ri[j] + ir[j];
    }
  }
}

// ---------------------------------------------------------------------------
// Kernel 1: per-head discretization.
// ---------------------------------------------------------------------------
__global__ void s4_discretize_kernel(
    const float* __restrict__ Lre, const float* __restrict__ Lim,
    const float* __restrict__ Pre, const float* __restrict__ Pim,
    const float* __restrict__ Bre, const float* __restrict__ Bim,
    const float* __restrict__ C,   const float* __restrict__ logstep,
    float* __restrict__ g_Ab, float* __restrict__ g_M16,
    float* __restrict__ g_Bb, float* __restrict__ g_Cb) {
  const int h   = blockIdx.x;
  const int tid = threadIdx.x;
  extern __shared__ float smem[];
  float* bufRe[3];
  float* bufIm[3];
  for (int i = 0; i < 3; ++i) {
    bufRe[i] = smem + (size_t)(2 * i) * 64 * MPAD;
    bufIm[i] = smem + (size_t)(2 * i + 1) * 64 * MPAD;
  }
  float* vec   = smem + (size_t)6 * 64 * MPAD;
  float* lamre = vec;        float* lamim = vec + 64;
  float* pre   = vec + 128;  float* pim   = vec + 192;
  float* ddre  = vec + 256;  float* ddim  = vec + 320;
  float* bvre  = vec + 384;  float* bvim  = vec + 448;
  float* ctre  = vec + 512;  float* ctim  = vec + 576;
  float* fre   = vec + 640;  float* fim   = vec + 704;
  float* scal  = vec + 768;

  const float step = __expf(logstep[h]);

  if (tid < 64) {
    const int n = tid;
    float lr = fminf(Lre[h * 64 + n], -1e-4f);
    float li = Lim[h * 64 + n];
    lamre[n] = lr; lamim[n] = li;
    pre[n] = Pre[h * 64 + n];  pim[n] = Pim[h * 64 + n];
    bvre[n] = Bre[h * 64 + n]; bvim[n] = Bim[h * 64 + n];
    ctre[n] = C[(h * 64 + n) * 2 + 0];
    ctim[n] = C[(h * 64 + n) * 2 + 1];
    float dr = 2.0f / step - lr, di = -li;      // Dd = 1/(2/step - Lam)
    float inv = 1.0f / (dr * dr + di * di);
    ddre[n] = dr * inv; ddim[n] = -di * inv;
  }
  __syncthreads();
  if (tid == 0) {
    float sr = 1.0f, si = 0.0f;                 // c = 1/(1 + sum |p|^2 Dd)
    for (int n = 0; n < 64; ++n) {
      float m = pre[n] * pre[n] + pim[n] * pim[n];
      sr += m * ddre[n];
      si += m * ddim[n];
    }
    float inv = 1.0f / (sr * sr + si * si);
    scal[0] = sr * inv;
    scal[1] = -si * inv;
  }
  __syncthreads();

  const float cre = scal[0], cim = scal[1];
  for (int idx = tid; idx < 4096; idx += 256) {
    const int i = idx >> 6, j = idx & 63;
    float ur = ddre[i] * pre[i] - ddim[i] * pim[i];
    float ui = ddre[i] * pim[i] + ddim[i] * pre[i];
    float vr = pre[j] * ddre[j] + pim[j] * ddim[j];
    float vi = pre[j] * ddim[j] - pim[j] * ddre[j];
    float tr = cre * vr - cim * vi;
    float ti = cre * vi + cim * vr;
    float a1r = -(ur * tr - ui * ti);
    float a1i = -(ur * ti + ui * tr);
    if (i == j) { a1r += ddre[i]; a1i += ddim[i]; }
    bufRe[0][i * MPAD + j] = a1r;
    bufIm[0][i * MPAD + j] = a1i;
    float a0r = -(pre[i] * pre[j] + pim[i] * pim[j]);
    float a0i = -(pim[i] * pre[j] - pre[i] * pim[j]);
    if (i == j) { a0r += 2.0f / step + lamre[i]; a0i += lamim[i]; }
    bufRe[1][i * MPAD + j] = a0r;
    bufIm[1][i * MPAD + j] = a0i;
  }
  __syncthreads();

  cmm64(bufRe[2], bufIm[2], bufRe[0], bufIm[0], bufRe[1], bufIm[1]);  // Ab
  __syncthreads();

  if (tid < 64) {                                // Bb = 2 * A1 @ B
    float ar = 0.f, ai = 0.f;
    for (int j = 0; j < 64; ++j) {
      float xr = bufRe[0][tid * MPAD + j], xi = bufIm[0][tid * MPAD + j];
      ar += xr * bvre[j] - xi * bvim[j];
      ai += xr * bvim[j] + xi * bvre[j];
    }
    g_Bb[h * 128 + tid]      = 2.f * ar;
    g_Bb[h * 128 + 64 + tid] = 2.f * ai;
  }
  for (int idx = tid; idx < 4096; idx += 256) {
    const int i = idx >> 6, j = idx & 63;
    g_Ab[(size_t)h * 8192 + idx]        = bufRe[2][i * MPAD + j];
    g_Ab[(size_t)h * 8192 + 4096 + idx] = bufIm[2][i * MPAD + j];
  }
  __syncthreads();

  int cur = 2;                                   // Ab^2048 by 11 squarings
  for (int it = 1; it <= 11; ++it) {
    const int dst = (cur == 2) ? 0 : 2;
    cmm64(bufRe[dst], bufIm[dst], bufRe[cur], bufIm[cur],
          bufRe[cur], bufIm[cur]);
    __syncthreads();
    cur = dst;
    if (it == 4) {                               // save Ab^16
      for (int idx = tid; idx < 4096; idx += 256) {
        const int i = idx >> 6, j = idx & 63;
        g_M16[(size_t)h * 8192 + idx]        = bufRe[cur][i * MPAD + j];
        g_M16[(size_t)h * 8192 + 4096 + idx] = bufIm[cur][i * MPAD + j];
      }
      __syncthreads();
    }
  }
  float* Wre = bufRe[1]; float* Wim = bufIm[1];  // W = I - AbL
  float* Vre = bufRe[2]; float* Vim = bufIm[2];  // Winv = I
  for (int idx = tid; idx < 4096; idx += 256) {
    const int i = idx >> 6, j = idx & 63;
    const float d = (i == j) ? 1.0f : 0.0f;
    Wre[i * MPAD + j] = d - bufRe[0][i * MPAD + j];
    Wim[i * MPAD + j] =   - bufIm[0][i * MPAD + j];
    Vre[i * MPAD + j] = d;
    Vim[i * MPAD + j] = 0.f;
  }
  __syncthreads();

  for (int k = 0; k < 64; ++k) {                 // complex Gauss-Jordan
    if (tid == 0) {
      float ar = Wre[k * MPAD + k], ai = Wim[k * MPAD + k];
      float inv = 1.0f / (ar * ar + ai * ai);
      scal[2] = ar * inv;
      scal[3] = -ai * inv;
    }
    __syncthreads();
    const float qr = scal[2], qi = scal[3];
    if (tid < 64) {
      const int j = tid;
      float wr = Wre[k * MPAD + j], wi = Wim[k * MPAD + j];
      Wre[k * MPAD + j] = wr * qr - wi * qi;
      Wim[k * MPAD + j] = wr * qi + wi * qr;
      float vr = Vre[k * MPAD + j], vi = Vim[k * MPAD + j];
      Vre[k * MPAD + j] = vr * qr - vi * qi;
      Vim[k * MPAD + j] = vr * qi + vi * qr;
    }
    __syncthreads();
    if (tid < 64) { fre[tid] = Wre[tid * MPAD + k]; fim[tid] = Wim[tid * MPAD + k]; }
    __syncthreads();
    for (int idx = tid; idx < 4096; idx += 256) {
      const int i = idx >> 6, j = idx & 63;
      if (i == k) continue;
      const float fr = fre[i], fi = fim[i];
      float wr = Wre[k * MPAD + j], wi = Wim[k * MPAD + j];
      Wre[i * MPAD + j] -= fr * wr - fi * wi;
      Wim[i * MPAD + j] -= fr * wi + fi * wr;
      float vr = Vre[k * MPAD + j], vi = Vim[k * MPAD + j];
      Vre[i * MPAD + j] -= fr * vr - fi * vi;
      Vim[i * MPAD + j] -= fr * vi + fi * vr;
    }
    __syncthreads();
  }

  if (tid < 64) {                                // Cb = conj(Ct) @ Winv
    const int k = tid;
    float ar = 0.f, ai = 0.f;
    for (int j = 0; j < 64; ++j) {
      float cr = ctre[j], ci = -ctim[j];
      float vr = Vre[j * MPAD + k], vi = Vim[j * MPAD + k];
      ar += cr * vr - ci * vi;
      ai += cr * vi + ci * vr;
    }
    g_Cb[h * 128 + k]      = ar;
    g_Cb[h * 128 + 64 + k] = ai;
  }
}

// ---------------------------------------------------------------------------
// Kernel 2: per-head SSM kernel generation. TDM stages Ab / Ab^16 with the
// hardware pad feature writing the MPAD=65 pitch directly.
// ---------------------------------------------------------------------------
__global__ void s4_kernelgen_kernel(const float* __restrict__ g_Ab,
                                    const float* __restrict__ g_M16,
                                    const float* __restrict__ g_Bb,
                                    const float* __restrict__ g_Cb,
                                    float* __restrict__ g_K) {
  const int h    = blockIdx.x;
  const int tid  = threadIdx.x;
  const int lane = tid & 31, wave = tid >> 5;
  const int lo   = lane & 15, hi = lane >> 4;
  extern __shared__ float smem[];
  float* Mre  = smem;
  float* Mim  = Mre + 64 * MPAD;
  float* curR = Mim + 64 * MPAD;     // 64 x 16, pitch 17
  float* curI = curR + 64 * 17;
  float* nxtR = curI + 64 * 17;
  float* nxtI = nxtR + 64 * 17;
  float* cbre = nxtI + 64 * 17;
  float* cbim = cbre + 64;

#if HAVE_TDM
  if (wave == 0) {
    // One DMA stages both planes; pad(64,1) => contiguous MPAD-pitched rows,
    // plane 1 lands exactly at Mre + 64*MPAD == Mim.
    tdm_load_2d((unsigned)(unsigned long long)Mre, g_Ab + (size_t)h * 8192,
                8192, 8192, 1, 1, 8192ull, /*pad*/1, /*every 64 dw*/5);
    __builtin_amdgcn_s_wait_tensorcnt((short)0);
  }
#else
  for (int idx = tid; idx < 4096; idx += 256) {
    const int i = idx >> 6, j = idx & 63;
    Mre[i * MPAD + j] = g_Ab[(size_t)h * 8192 + idx];
    Mim[i * MPAD + j] = g_Ab[(size_t)h * 8192 + 4096 + idx];
  }
#endif
  __builtin_prefetch(g_M16 + (size_t)h * 8192, 0, 1);   // global_prefetch_b8
  if (tid < 64) {
    curR[tid * 17 + 0] = g_Bb[h * 128 + tid];
    curI[tid * 17 + 0] = g_Bb[h * 128 + 64 + tid];
    cbre[tid] = g_Cb[h * 128 + tid];
    cbim[tid] = g_Cb[h * 128 + 64 + tid];
  }
  __syncthreads();

  for (int j = 1; j < 16; ++j) {                 // first 16 columns b_j
    if (tid < 64) {
      float ar = 0.f, ai = 0.f;
      for (int n = 0; n < 64; ++n) {
        float mr = Mre[tid * MPAD + n], mi = Mim[tid * MPAD + n];
        float xr = curR[n * 17 + j - 1], xi = curI[n * 17 + j - 1];
        ar += mr * xr - mi * xi;
        ai += mr * xi + mi * xr;
      }
      curR[tid * 17 + j] = ar;
      curI[tid * 17 + j] = ai;
    }
    __syncthreads();
  }
  if (tid < 16) {
    float acc = 0.f;
    for (int n = 0; n < 64; ++n)
      acc += cbre[n] * curR[n * 17 + tid] - cbim[n] * curI[n * 17 + tid];
    g_K[(size_t)h * L_DIM + tid] = acc;
  }
  __syncthreads();

#if HAVE_TDM
  if (wave == 0) {
    tdm_load_2d((unsigned)(unsigned long long)Mre, g_M16 + (size_t)h * 8192,
                8192, 8192, 1, 1, 8192ull, 1, 5);
    __builtin_amdgcn_s_wait_tensorcnt((short)0);
  }
#else
  for (int idx = tid; idx < 4096; idx += 256) {
    const int i = idx >> 6, j = idx & 63;
    Mre[i * MPAD + j] = g_M16[(size_t)h * 8192 + idx];
    Mim[i * MPAD + j] = g_M16[(size_t)h * 8192 + 4096 + idx];
  }
#endif
  __syncthreads();

  for (int c = 1; c < 128; ++c) {
    const int  mi     = wave & 3;
    const bool doReal = wave < 4;
    v8f acc0 = {}, acc1 = {};
    for (int kk = 0; kk < 16; ++kk) {
      v2f aRe, aIm, bRe, bIm;
      const int arow = 16 * mi + lo;
      const int acol = 4 * kk + 2 * hi;
      aRe.x = Mre[arow * MPAD + acol]; aRe.y = Mre[arow * MPAD + acol + 1];
      aIm.x = Mim[arow * MPAD + acol]; aIm.y = Mim[arow * MPAD + acol + 1];
      const int brow = 4 * kk + 2 * hi;
      bRe.x = curR[brow * 17 + lo];    bRe.y = curR[(brow + 1) * 17 + lo];
      bIm.x = curI[brow * 17 + lo];    bIm.y = curI[(brow + 1) * 17 + lo];
      if (doReal) {
        acc0 = wmma_f32x4(aRe, bRe, acc0);
        acc1 = wmma_f32x4(aIm, bIm, acc1);
      } else {
        acc0 = wmma_f32x4(aRe, bIm, acc0);
        acc1 = wmma_f32x4(aIm, bRe, acc1);
      }
    }
    for (int j = 0; j < 8; ++j) {
      const int row = 16 * mi + j + 8 * hi;
      if (doReal) nxtR[row * 17 + lo] = acc0[j] - acc1[j];
      else        nxtI[row * 17 + lo] = acc0[j] + acc1[j];
    }
    __syncthreads();
    if (tid < 16) {
      float acc = 0.f;
      for (int n = 0; n < 64; ++n)
        acc += cbre[n] * nxtR[n * 17 + tid] - cbim[n] * nxtI[n * 17 + tid];
      g_K[(size_t)h * L_DIM + 16 * c + tid] = acc;
    }
    float* t;
    t = curR; curR = nxtR; nxtR = t;
    t = curI; curI = nxtI; nxtI = t;
    __syncthreads();
  }
}

// ---------------------------------------------------------------------------
// Kernel 3: causal Toeplitz convolution y = K * u + D*u via WMMA.
// TDM stages the contiguous K row and the H-strided u column into LDS.
// ---------------------------------------------------------------------------
__global__ void s4_conv_kernel(const float* __restrict__ g_K,
                               const float* __restrict__ u,
                               const float* __restrict__ Dvec,
                               float* __restrict__ out) {
  const int h    = blockIdx.x;
  const int tid  = threadIdx.x;
  const int lane = tid & 31, wave = tid >> 5;
  const int lo   = lane & 15, hi = lane >> 4;
  extern __shared__ float smem[];
  float* Kl = smem;         // [2048]
  float* ul = Kl + L_DIM;   // [2048]

#if HAVE_TDM
  if (wave == 0) {
    tdm_load_2d((unsigned)(unsigned long long)Kl, g_K + (size_t)h * L_DIM,
                L_DIM, L_DIM, 1, 1, (unsigned long long)L_DIM, 0, 0);
    // u[t*H + h], t = 0..L-1: 2D tile 1 x L with row stride H (element units)
    tdm_load_2d((unsigned)(unsigned long long)ul, u + h,
                1, 1, L_DIM, L_DIM, (unsigned long long)H_DIM, 0, 0);
    __builtin_amdgcn_s_wait_tensorcnt((short)0);
  }
#else
  for (int t = tid; t < L_DIM; t += 256) {
    Kl[t] = g_K[(size_t)h * L_DIM + t];
    ul[t] = u[(size_t)t * H_DIM + h];
  }
#endif
  __syncthreads();

  const float dval = Dvec[h];
  const int   tiLo = 16 * wave;
  v8f acc = {};
  for (int dti = 0; dti <= tiLo + 15; ++dti) {
    for (int kk = 0; kk < 4; ++kk) {
      v2f A, B;
      const int a    = lo;
      const int b0   = 4 * kk + 2 * hi;
      const int idx0 = 16 * dti + a - b0;
      A.x = (idx0 >= 0) ? Kl[idx0] : 0.f;
      A.y = (idx0 - 1 >= 0) ? Kl[idx0 - 1] : 0.f;
      const int  ti = tiLo + lo;
      const bool v  = (ti >= dti);
      const int  s0 = 16 * (ti - dti) + b0;
      B.x = v ? ul[s0] : 0.f;
      B.y = v ? ul[s0 + 1] : 0.f;
      acc = wmma_f32x4(A, B, acc);
    }
  }
  for (int j = 0; j < 8; ++j) {
    const int a  = j + 8 * hi;
    const int ti = tiLo + lo;
    const int t  = 16 * ti + a;
    out[(size_t)t * H_DIM + h] = acc[j] + dval * ul[t];
  }
}

// ---------------------------------------------------------------------------
extern "C" void kernel_launch(void* const* d_in, const int* in_sizes, int n_in,
                              void* d_out, int out_size, void* d_ws,
                              size_t ws_size, hipStream_t stream) {
  const float* u     = (const float*)d_in[0];
  // d_in[1], d_in[2] = x0_re/x0_im, zero by construction (conv form assumes 0)
  const float* Lre   = (const float*)d_in[3];
  const float* Lim   = (const float*)d_in[4];
  const float* Pre   = (const float*)d_in[5];
  const float* Pim   = (const float*)d_in[6];
  const float* Bre   = (const float*)d_in[7];
  const float* Bim   = (const float*)d_in[8];
  const float* C     = (const float*)d_in[9];
  const float* Dv    = (const float*)d_in[10];
  const float* lstep = (const float*)d_in[11];

  float* ws    = (float*)d_ws;
  float* g_K   = ws;                                        // H*L
  float* g_Ab  = g_K  + (size_t)H_DIM * L_DIM;              // H*2*N*N
  float* g_M16 = g_Ab + (size_t)H_DIM * 2 * N_DIM * N_DIM;  // H*2*N*N
  float* g_Bb  = g_M16 + (size_t)H_DIM * 2 * N_DIM * N_DIM; // H*2*N
  float* g_Cb  = g_Bb + (size_t)H_DIM * 2 * N_DIM;          // H*2*N

  const size_t smem1 = (size_t)(6 * 64 * MPAD + 1024) * sizeof(float); // ~104KB
  const size_t smem2 = (size_t)(2 * 64 * MPAD + 4 * 64 * 17 + 128) * sizeof(float);
  const size_t smem3 = (size_t)(2 * L_DIM) * sizeof(float);

  s4_discretize_kernel<<<H_DIM, 256, smem1, stream>>>(
      Lre, Lim, Pre, Pim, Bre, Bim, C, lstep, g_Ab, g_M16, g_Bb, g_Cb);
  s4_kernelgen_kernel<<<H_DIM, 256, smem2, stream>>>(
      g_Ab, g_M16, g_Bb, g_Cb, g_K);
  s4_conv_kernel<<<H_DIM, 256, smem3, stream>>>(g_K, u, Dv, (float*)d_out);
}